// SessionGraph_46540265619956
// MI455X (gfx1250) — compile-verified
//
#include <hip/hip_runtime.h>

typedef float v2f __attribute__((ext_vector_type(2)));
typedef float v8f __attribute__((ext_vector_type(8)));
typedef int   v4i __attribute__((ext_vector_type(4)));

#define BSZ   256
#define NN    50
#define DD    128
#define KATTR 2000
#define HS    129   // h_s row stride (floats): (j*129+d)%64 distinct across 32 lanes
#define AS    68    // alpha_s row stride: (m*68+k)%64 = (4m+k)%64 distinct for m<16
#define ASTR  104   // attr A_s row stride: 16B-aligned for async b128 (2-way conflict ok)
#define NEGV  (-9e15f)

#if defined(__gfx1250__) && __has_builtin(__builtin_amdgcn_global_load_async_to_lds_b128)
#define HAVE_ASYNC_LDS 1
#else
#define HAVE_ASYNC_LDS 0
#endif

// ---------------------------------------------------------------------------
// Kernel 1: one block per batch. Gather h, masked-attention logits + softmax,
// then output = alpha @ h via V_WMMA_F32_16X16X4_F32 (K padded 50 -> 64).
// M-tiles {0,16,32,34} overlap so every output row is valid -> unguarded
// stores (no saveexec chains), no wasted WMMA on pad rows.
// ---------------------------------------------------------------------------
__global__ __launch_bounds__(256) void session_graph_kernel(
    const int*   __restrict__ inputs,     // [B,N]
    const int*   __restrict__ adj,        // [B,N,N]
    const float* __restrict__ embedding,  // [N_NODE,D]
    const float* __restrict__ a0, const float* __restrict__ a1,
    const float* __restrict__ a2, const float* __restrict__ a3,
    float*       __restrict__ out)        // [B,N,D]
{
    __shared__ float h_s[64 * HS];       // rows 50..63 zeroed (K padding)
    __shared__ float alpha_s[NN * AS];   // 50 rows x 64 cols (cols >=50 zeroed)
    __shared__ float a_lds[4 * HS];      // a0..a3, stride 129

    const int b    = blockIdx.x;
    const int tid  = threadIdx.x;
    const int lane = tid & 31;
    const int wave = tid >> 5;

    // ---- Phase 1: gather h, zero K-pad rows, stage a0..a3 ----
    for (int idx = tid; idx < NN * DD; idx += 256) {
        int n = idx >> 7, d = idx & 127;
        h_s[n * HS + d] = embedding[(long)inputs[b * NN + n] * DD + d];
    }
    for (int idx = tid; idx < (64 - NN) * DD; idx += 256) {
        int n = NN + (idx >> 7), d = idx & 127;
        h_s[n * HS + d] = 0.0f;
    }
    for (int idx = tid; idx < 4 * DD; idx += 256) {
        int k = idx >> 7, d = idx & 127;
        const float* ap = (k == 0) ? a0 : (k == 1) ? a1 : (k == 2) ? a2 : a3;
        a_lds[k * HS + d] = ap[d];
    }
    __syncthreads();

    // ---- Phase 2: logits (adj selects ONE of 4 dot products) + softmax ----
    const int* adjb = adj + b * NN * NN;
    for (int i = wave; i < NN; i += 8) {
        const float* hi = &h_s[i * HS];
        float s0 = NEGV, s1 = NEGV;
        #pragma unroll
        for (int half = 0; half < 2; ++half) {
            int j = lane + 32 * half;
            float sv = NEGV;
            if (j < NN) {
                int av = adjb[i * NN + j];                  // coalesced over lanes
                if ((unsigned)(av - 1) < 4u) {
                    const float* asel = &a_lds[(av - 1) * HS];
                    const float* hj   = &h_s[j * HS];       // conflict-free (stride 129)
                    float acc = 0.0f;
                    #pragma unroll 4
                    for (int d = 0; d < DD; ++d) {
                        float p = hi[d] * hj[d];            // hi[d] is an LDS broadcast
                        p = (p > 0.0f) ? p : 0.2f * p;      // LeakyReLU(0.2)
                        acc = fmaf(p, asel[d], acc);
                    }
                    sv = acc;
                }
            }
            if (half == 0) s0 = sv; else s1 = sv;
        }
        // wave-wide softmax over the 50 logits of row i
        float m = fmaxf(s0, s1);
        for (int off = 16; off > 0; off >>= 1) m = fmaxf(m, __shfl_xor(m, off, 32));
        float e0 = __expf(s0 - m);
        float e1 = (lane + 32 < NN) ? __expf(s1 - m) : 0.0f;
        float sum = e0 + e1;
        for (int off = 16; off > 0; off >>= 1) sum += __shfl_xor(sum, off, 32);
        float inv = 1.0f / sum;   // all-masked row: every logit == NEGV -> uniform, as in ref
        alpha_s[i * AS + lane] = e0 * inv;
        alpha_s[i * AS + lane + 32] = (lane + 32 < NN) ? e1 * inv : 0.0f; // zero K-pad cols
    }
    __syncthreads();

    // ---- Phase 3: output = alpha(50x50, K pad 64) @ h(50x128) via WMMA f32 ----
    const int mrow = lane & 15;
    const int kk   = (lane >> 4) << 1;   // hi half-wave covers K+2,K+3
    float* outb = out + (long)b * NN * DD;
    for (int t = wave; t < 32; t += 8) { // 4 M-tiles x 8 N-tiles
        int mi = t & 3;
        int m0 = (mi == 3) ? 34 : mi * 16;   // overlapping last tile: rows 34..49
        int n0 = (t >> 2) * 16;
        v8f c = {};
        #pragma unroll 4
        for (int k = 0; k < 64; k += 4) {
            v2f a, bb;
            a.x  = alpha_s[(m0 + mrow) * AS + k + kk];
            a.y  = alpha_s[(m0 + mrow) * AS + k + kk + 1];
            bb.x = h_s[(k + kk) * HS + n0 + mrow];
            bb.y = h_s[(k + kk + 1) * HS + n0 + mrow];
            c = __builtin_amdgcn_wmma_f32_16x16x4_f32(false, a, false, bb,
                                                      (short)0, c, false, false);
        }
        int rbase = m0 + ((lane >> 4) << 3);
        #pragma unroll
        for (int r = 0; r < 8; ++r)          // all rows valid -> no exec masking
            outb[(rbase + r) * DD + n0 + mrow] = c[r];
    }
}

// ---------------------------------------------------------------------------
// Kernel 2: attr_sess = A_attr_sess[256,2000] @ attr_embedding[2000,128].
// 16 blocks x 8 waves; each wave owns one 16x16 C tile. A chunks (16x100)
// double-buffered in LDS via GLOBAL_LOAD_ASYNC_TO_LDS_B128 (fallback: sync).
// ---------------------------------------------------------------------------
#define KC     100          // K-chunk (multiple of 4; 100*4B keeps 16B alignment)
#define NCHUNK (KATTR / KC) // 20

__device__ __forceinline__ void stage_chunk(const float* __restrict__ A,
                                            float* __restrict__ dst,
                                            int m0, int kc, int tid)
{
    // 16 rows x 25 float4 = 400 b128 transfers, spread over 256 threads
    for (int idx = tid; idx < 16 * 25; idx += 256) {
        int r = idx / 25, q = idx - r * 25;
        const float* src = A + (m0 + r) * KATTR + kc + q * 4;   // 16B aligned
        float*       d   = dst + r * ASTR + q * 4;              // 16B aligned
#if HAVE_ASYNC_LDS
        __builtin_amdgcn_global_load_async_to_lds_b128(
            (__attribute__((address_space(1))) v4i*)src,
            (__attribute__((address_space(3))) v4i*)d, 0, 0);
#else
        d[0] = src[0]; d[1] = src[1]; d[2] = src[2]; d[3] = src[3];
#endif
    }
}

__device__ __forceinline__ void wait_async_lds()
{
#if HAVE_ASYNC_LDS
#if __has_builtin(__builtin_amdgcn_s_wait_asynccnt)
    __builtin_amdgcn_s_wait_asynccnt(0);
#else
    asm volatile("s_wait_asynccnt 0x0" ::: "memory");
#endif
#endif
}

__global__ __launch_bounds__(256) void attr_gemm_kernel(
    const float* __restrict__ A,   // [256,2000]
    const float* __restrict__ Bm,  // [2000,128]
    float*       __restrict__ C)   // [256,128]
{
    __shared__ __align__(16) float A_s[2][16 * ASTR];
    const int tid  = threadIdx.x;
    const int lane = tid & 31;
    const int wave = tid >> 5;
    const int m0   = blockIdx.x * 16;
    const int n0   = wave * 16;
    const int mrow = lane & 15;
    const int kk   = (lane >> 4) << 1;

    stage_chunk(A, A_s[0], m0, 0, tid);          // prefetch chunk 0

    v8f c = {};
    for (int ci = 0; ci < NCHUNK; ++ci) {
        wait_async_lds();        // my loads for chunk ci have landed in LDS
        __syncthreads();         // everyone's loads landed; prev buffer fully consumed
        if (ci + 1 < NCHUNK)     // overlap: stage next chunk while we compute
            stage_chunk(A, A_s[(ci + 1) & 1], m0, (ci + 1) * KC, tid);

        const float* As = A_s[ci & 1];
        const int    kc = ci * KC;
        #pragma unroll 5
        for (int k = 0; k < KC; k += 4) {
            v2f a, bb;
            a.x  = As[mrow * ASTR + k + kk];
            a.y  = As[mrow * ASTR + k + kk + 1];
            bb.x = Bm[(kc + k + kk) * DD + n0 + mrow];      // coalesced over lanes
            bb.y = Bm[(kc + k + kk + 1) * DD + n0 + mrow];
            c = __builtin_amdgcn_wmma_f32_16x16x4_f32(false, a, false, bb,
                                                      (short)0, c, false, false);
        }
    }
    int rbase = m0 + ((lane >> 4) << 3);
    #pragma unroll
    for (int r = 0; r < 8; ++r)
        C[(rbase + r) * DD + n0 + mrow] = c[r];
}

extern "C" void kernel_launch(void* const* d_in, const int* in_sizes, int n_in,
                              void* d_out, int out_size, void* d_ws, size_t ws_size,
                              hipStream_t stream) {
    (void)in_sizes; (void)n_in; (void)out_size; (void)d_ws; (void)ws_size;
    const int*   inputs    = (const int*)d_in[0];
    const int*   adj       = (const int*)d_in[1];
    /* d_in[2] = mask_item: unused by the reference */
    const float* A_attr    = (const float*)d_in[3];
    const float* embedding = (const float*)d_in[4];
    const float* attr_emb  = (const float*)d_in[5];
    const float* a0        = (const float*)d_in[6];
    const float* a1        = (const float*)d_in[7];
    const float* a2        = (const float*)d_in[8];
    const float* a3        = (const float*)d_in[9];

    float* out      = (float*)d_out;                 // [256,50,128]
    float* attr_out = out + (long)BSZ * NN * DD;     // [256,128]

    session_graph_kernel<<<BSZ, 256, 0, stream>>>(inputs, adj, embedding,
                                                  a0, a1, a2, a3, out);
    attr_gemm_kernel<<<16, 256, 0, stream>>>(A_attr, attr_emb, attr_out);
}